// PITLoss_25512105738972
// MI455X (gfx1250) — compile-verified
//
#include <hip/hip_runtime.h>

// PIT loss for [B=8, C=4, CH=2, T=262144] fp32.
// Memory-bound: 128 MiB streamed once -> ~5.5us floor at 23.3 TB/s.
// Stage 1: per-batch Gram terms (sx, sy, cross) via V_WMMA_F32_16X16X4_F32,
//          each element loaded exactly once as a b64 per lane.
// Stage 2: tiny kernel enumerates 24 permutations per batch, emits total + assignments.

typedef __attribute__((ext_vector_type(2))) float v2f;
typedef __attribute__((ext_vector_type(8))) float v8f;

#define NB 8
#define NC 4
#define ND 524288          // CH * T, contiguous per (b,c) row
#define KCHUNK 1024        // K columns handled per block (1024 blocks_x*2 tiles)

// ws layout (floats): cross[2][16][16] | sx[32] | sy[32]  => 576 floats
#define WS_FLOATS 576

__global__ __launch_bounds__(32)
void pit_partial_wmma(const float* __restrict__ x,
                      const float* __restrict__ y,
                      float* __restrict__ ws) {
    const int lane  = threadIdx.x;        // wave32: one wave per block
    const int tile  = blockIdx.y;         // 0: rows 0..15 (batches 0..3); 1: rows 16..31
    const int row   = lane & 15;          // M (A) and N (B) index of this lane
    const int khalf = lane >> 4;          // 0: K={0,1}, 1: K={2,3}
    const size_t r  = (size_t)tile * 16 + row;          // flat (b*4 + c) row
    const size_t k0 = (size_t)blockIdx.x * KCHUNK;

    // A = Y rows (M=(b,i)), B = X rows as columns (N=(b,j)).
    // Both layouts want a contiguous float2 per lane per K-step.
    const float* ya = y + r * (size_t)ND + k0 + (size_t)(khalf * 2);
    const float* xb = x + r * (size_t)ND + k0 + (size_t)(khalf * 2);

    v8f  acc   = {};      // 16x16 fp32 accumulator: D[M,N] = sum_K Y[M,K]*X[N,K]
    float sxacc = 0.0f;   // sum x^2 for column N=row over this lane's K subset
    float syacc = 0.0f;   // sum y^2 for row    M=row over this lane's K subset

    #pragma unroll 8
    for (int k = 0; k < KCHUNK; k += 4) {
        v2f a = *(const v2f*)(ya + k);    // y: A-matrix fragment (K even/odd pair)
        v2f b = *(const v2f*)(xb + k);    // x: B-matrix fragment
        acc = __builtin_amdgcn_wmma_f32_16x16x4_f32(
                  /*neg_a=*/false, a, /*neg_b=*/false, b,
                  /*c_mod=*/(short)0, acc, /*reuse_a=*/false, /*reuse_b=*/false);
        sxacc = __builtin_fmaf(b.x, b.x, sxacc);
        sxacc = __builtin_fmaf(b.y, b.y, sxacc);
        syacc = __builtin_fmaf(a.x, a.x, syacc);
        syacc = __builtin_fmaf(a.y, a.y, syacc);
    }

    float* cross = ws;          // [2][16][16]
    float* sx    = ws + 512;    // [32]
    float* sy    = ws + 544;    // [32]

    atomicAdd(&sx[tile * 16 + row], sxacc);
    atomicAdd(&sy[tile * 16 + row], syacc);

    // D layout: VGPR rr -> M = rr + khalf*8, N = lane&15. Only the per-batch
    // block-diagonal (M>>2 == N>>2) is part of the answer; skip the rest.
    #pragma unroll
    for (int rr = 0; rr < 8; ++rr) {
        const int M = rr + khalf * 8;
        const int N = row;
        if ((M >> 2) == (N >> 2)) {
            atomicAdd(&cross[tile * 256 + M * 16 + N], acc[rr]);
        }
    }
}

__global__ __launch_bounds__(32)
void pit_finalize(const float* __restrict__ ws, float* __restrict__ out) {
    // itertools.permutations(range(4)) in lexicographic order
    static __device__ const int perms[24][4] = {
        {0,1,2,3},{0,1,3,2},{0,2,1,3},{0,2,3,1},{0,3,1,2},{0,3,2,1},
        {1,0,2,3},{1,0,3,2},{1,2,0,3},{1,2,3,0},{1,3,0,2},{1,3,2,0},
        {2,0,1,3},{2,0,3,1},{2,1,0,3},{2,1,3,0},{2,3,0,1},{2,3,1,0},
        {3,0,1,2},{3,0,2,1},{3,1,0,2},{3,1,2,0},{3,2,0,1},{3,2,1,0}};

    __shared__ float mincost[NB];
    const int b = threadIdx.x;
    if (b < NB) {
        const float* cross = ws;
        const float* sx    = ws + 512;
        const float* sy    = ws + 544;
        const int tile = b >> 2;
        const float invD = 1.0f / (float)ND;

        float dists[4][4];
        for (int i = 0; i < 4; ++i) {
            for (int j = 0; j < 4; ++j) {
                const int M = (b * 4 + i) & 15;
                const int N = (b * 4 + j) & 15;
                dists[i][j] = (sy[b * 4 + i] + sx[b * 4 + j]
                               - 2.0f * cross[tile * 256 + M * 16 + N]) * invD;
            }
        }
        float best = 3.402823e38f;
        int bestp = 0;
        for (int p = 0; p < 24; ++p) {
            const float c = dists[0][perms[p][0]] + dists[1][perms[p][1]]
                          + dists[2][perms[p][2]] + dists[3][perms[p][3]];
            if (c < best) { best = c; bestp = p; }
        }
        mincost[b] = best;
        for (int i = 0; i < 4; ++i)
            out[1 + b * 4 + i] = (float)perms[bestp][i];
    }
    __syncthreads();
    if (b == 0) {
        float t = 0.0f;
        for (int i = 0; i < NB; ++i) t += mincost[i];
        out[0] = t;   // total PIT loss, followed by assignments [8][4]
    }
}

extern "C" void kernel_launch(void* const* d_in, const int* in_sizes, int n_in,
                              void* d_out, int out_size, void* d_ws, size_t ws_size,
                              hipStream_t stream) {
    const float* x = (const float*)d_in[0];
    const float* y = (const float*)d_in[1];
    float* out = (float*)d_out;
    float* ws  = (float*)d_ws;

    // Zero the 2.3 KiB accumulator region each launch (graph-capture safe).
    hipMemsetAsync(d_ws, 0, WS_FLOATS * sizeof(float), stream);

    dim3 grid(ND / KCHUNK, 2);   // 512 x 2 = 1024 single-wave blocks
    dim3 block(32);
    pit_partial_wmma<<<grid, block, 0, stream>>>(x, y, ws);
    pit_finalize<<<1, 32, 0, stream>>>(ws, out);
}